// postprocess_ssfd_43920335569671
// MI455X (gfx1250) — compile-verified
//
#include <hip/hip_runtime.h>
#include <math.h>
#include <stdint.h>

// ---------------- problem constants (from reference) ----------------
#define NDET   21840            // total anchors: 128^2+64^2+32^2+16^2+8^2+4^2
#define NGRP   683              // ceil(NDET/32)
#define NPAD   (NGRP*32)        // 21856
#define MSORT  32768            // power-of-two pad for bitonic sort
#define TILE   2048             // LDS tile for fused bitonic stages
#define STHREADS 1024
#define CLS_THf 0.05f
#define NMS_THf 0.3f
#define FIN_THf 0.5f
#define VAR0f   0.1f
#define VAR1f   0.2f
#define BIGKEY  3.402823466e38f

struct InPtrs { const float* p[12]; };

__device__ __forceinline__ void map_scale(int n, int& si, int& hw, int& stride, int& base) {
  if      (n < 16384) { si = 0; hw = 128; stride = 4;   base = 0;     }
  else if (n < 20480) { si = 1; hw = 64;  stride = 8;   base = 16384; }
  else if (n < 21504) { si = 2; hw = 32;  stride = 16;  base = 20480; }
  else if (n < 21760) { si = 3; hw = 16;  stride = 32;  base = 21504; }
  else if (n < 21824) { si = 4; hw = 8;   stride = 64;  base = 21760; }
  else                { si = 5; hw = 4;   stride = 128; base = 21824; }
}

// ---------------- 1) decode + sort-key init ----------------
__global__ void decode_init_kernel(InPtrs in,
                                   float* __restrict__ dx1, float* __restrict__ dy1,
                                   float* __restrict__ dx2, float* __restrict__ dy2,
                                   float* __restrict__ dsc, int* __restrict__ keep,
                                   float* __restrict__ key, int* __restrict__ idx) {
  int n = blockIdx.x * blockDim.x + threadIdx.x;
  if (n >= MSORT) return;
  if (n < NDET) {
    int si, hw, stride, base; map_scale(n, si, hw, stride, base);
    int r   = n - base;
    int y   = r / hw;
    int x   = r - y * hw;
    int hw2 = hw * hw;
    const float* cls = in.p[2 * si];
    const float* reg = in.p[2 * si + 1];
    float c0 = cls[r], c1 = cls[hw2 + r];
    float prob = 1.0f / (1.0f + expf(c0 - c1));     // softmax class-1
    float s   = (float)stride;
    float pwh = 4.0f * s;
    float cx  = (0.5f + (float)x) * s + reg[r]       * (VAR0f * pwh);
    float cy  = (0.5f + (float)y) * s + reg[hw2 + r] * (VAR0f * pwh);
    float w   = pwh * expf(reg[2 * hw2 + r] * VAR1f);
    float h   = pwh * expf(reg[3 * hw2 + r] * VAR1f);
    float X1 = cx - 0.5f * w, Y1 = cy - 0.5f * h;
    float X2 = X1 + w,        Y2 = Y1 + h;
    dx1[n] = X1; dy1[n] = Y1; dx2[n] = X2; dy2[n] = Y2; dsc[n] = prob;
    keep[n] = 0;
    key[n]  = (prob > CLS_THf) ? -prob : BIGKEY;   // ascending key = descending score
    idx[n]  = n;
  } else {
    if (n < NPAD) keep[n] = 0;
    key[n] = BIGKEY;
    idx[n] = n;                                    // >= NDET marks pad
  }
}

// ---------------- 2) bitonic sort ----------------
// Fused LDS kernel: full sort of 2048-element tiles, k = 2 .. TILE (all j local).
__global__ __launch_bounds__(1024) void bitonic_local_full_kernel(float* __restrict__ key,
                                                                  int* __restrict__ idx) {
  __shared__ float sk[TILE];
  __shared__ int   si[TILE];
  const int t    = threadIdx.x;
  const int base = blockIdx.x * TILE;
  sk[t] = key[base + t];                       si[t] = idx[base + t];
  sk[t + STHREADS] = key[base + t + STHREADS]; si[t + STHREADS] = idx[base + t + STHREADS];
  for (int k = 2; k <= TILE; k <<= 1) {
    for (int j = k >> 1; j >= 1; j >>= 1) {
      __syncthreads();
      int pos = ((t & ~(j - 1)) << 1) | (t & (j - 1));
      int par = pos | j;
      bool asc = (((base + pos) & k) == 0);
      float a = sk[pos], b = sk[par];
      if (asc ? (a > b) : (a < b)) {
        sk[pos] = b; sk[par] = a;
        int tmp = si[pos]; si[pos] = si[par]; si[par] = tmp;
      }
    }
  }
  __syncthreads();
  key[base + t] = sk[t];                       idx[base + t] = si[t];
  key[base + t + STHREADS] = sk[t + STHREADS]; idx[base + t + STHREADS] = si[t + STHREADS];
}

// Fused LDS kernel: stages j = TILE/2 .. 1 for a fixed k (direction uniform per tile).
__global__ __launch_bounds__(1024) void bitonic_local_merge_kernel(float* __restrict__ key,
                                                                   int* __restrict__ idx, int k) {
  __shared__ float sk[TILE];
  __shared__ int   si[TILE];
  const int t    = threadIdx.x;
  const int base = blockIdx.x * TILE;
  sk[t] = key[base + t];                       si[t] = idx[base + t];
  sk[t + STHREADS] = key[base + t + STHREADS]; si[t + STHREADS] = idx[base + t + STHREADS];
  for (int j = TILE >> 1; j >= 1; j >>= 1) {
    __syncthreads();
    int pos = ((t & ~(j - 1)) << 1) | (t & (j - 1));
    int par = pos | j;
    bool asc = (((base + pos) & k) == 0);
    float a = sk[pos], b = sk[par];
    if (asc ? (a > b) : (a < b)) {
      sk[pos] = b; sk[par] = a;
      int tmp = si[pos]; si[pos] = si[par]; si[par] = tmp;
    }
  }
  __syncthreads();
  key[base + t] = sk[t];                       idx[base + t] = si[t];
  key[base + t + STHREADS] = sk[t + STHREADS]; idx[base + t + STHREADS] = si[t + STHREADS];
}

// Global step kernel (only needed for j >= TILE).
__global__ void bitonic_step_kernel(float* __restrict__ key, int* __restrict__ idx, int j, int k) {
  int tid = blockIdx.x * blockDim.x + threadIdx.x;
  if (tid >= MSORT) return;
  int ixj = tid ^ j;
  if (ixj <= tid) return;
  bool ascend = ((tid & k) == 0);
  float a = key[tid], b = key[ixj];
  if (ascend ? (a > b) : (a < b)) {
    key[tid] = b; key[ixj] = a;
    int t = idx[tid]; idx[tid] = idx[ixj]; idx[ixj] = t;
  }
}

// ---------------- 3) gather boxes into sorted order (AoS float4 + area) ----------------
__global__ void gather_kernel(const float* __restrict__ dx1, const float* __restrict__ dy1,
                              const float* __restrict__ dx2, const float* __restrict__ dy2,
                              const float* __restrict__ key, const int* __restrict__ idx,
                              float4* __restrict__ bb, float* __restrict__ bar,
                              float* __restrict__ bvl) {
  int i = blockIdx.x * blockDim.x + threadIdx.x;
  if (i >= NPAD) return;
  int  id    = idx[i];
  bool valid = (id < NDET) && (key[i] < 0.0f);     // key<0 <=> score>CLS_TH
  if (valid) {
    float X1 = dx1[id], Y1 = dy1[id], X2 = dx2[id], Y2 = dy2[id];
    bb[i]  = make_float4(X1, Y1, X2, Y2);
    bar[i] = (X2 - X1 + 1.0f) * (Y2 - Y1 + 1.0f);
    bvl[i] = 1.0f;
  } else {
    bb[i]  = make_float4(0.0f, 0.0f, 0.0f, 0.0f);
    bar[i] = 1.0f;
    bvl[i] = 0.0f;
  }
}

// async global->LDS staging (CDNA5 path, ASYNCcnt tracked)
#define ASYNC_LD_B128(ldsp, gp)                                                   \
  asm volatile("global_load_async_to_lds_b128 %0, %1, off"                        \
               :: "v"((unsigned)(unsigned long long)(ldsp)),                      \
                  "v"((unsigned long long)(gp)) : "memory")
#define ASYNC_LD_B32(ldsp, gp)                                                    \
  asm volatile("global_load_async_to_lds_b32 %0, %1, off"                         \
               :: "v"((unsigned)(unsigned long long)(ldsp)),                      \
                  "v"((unsigned long long)(gp)) : "memory")

// ---------------- 4) exact greedy NMS, single workgroup ----------------
__global__ __launch_bounds__(1024) void nms_kernel(
    const float4* __restrict__ bb, const float* __restrict__ bar,
    const float* __restrict__ bvl, const int* __restrict__ sidx,
    int* __restrict__ keep) {
  __shared__ unsigned mask[NGRP];                  // "removed" bit per sorted position
  __shared__ float4 sbox[32];
  __shared__ float  sar[32], svl[32];
  __shared__ unsigned sActive;
  const int tid = threadIdx.x;

  for (int w = tid; w < NGRP; w += blockDim.x) mask[w] = 0u;
  __syncthreads();

  for (int g = 0; g < NGRP; ++g) {
    // --- stage this group's 32 boxes into LDS via the async-to-LDS path ---
    if (tid < 32) {
      int i = g * 32 + tid;
      ASYNC_LD_B128(&sbox[tid], bb + i);
      ASYNC_LD_B32(&sar[tid], bar + i);
      ASYNC_LD_B32(&svl[tid], bvl + i);
      asm volatile("s_wait_asynccnt 0" ::: "memory");
    }
    __syncthreads();

    // --- wave0: serial intra-group greedy resolution (wave32-native) ---
    if (tid < 32) {
      const int lane = tid;
      const int i    = g * 32 + lane;
      float4 B = sbox[lane];
      float X1 = B.x, Y1 = B.y, X2 = B.z, Y2 = B.w, AR = sar[lane];
      bool valid = svl[lane] > 0.5f;
      bool sup   = ((mask[g] >> lane) & 1u) != 0u;
      unsigned validMask  = (unsigned)__ballot(valid);
      unsigned activeMask = 0u;
      for (int l = 0; l < 32; ++l) {
        unsigned supb = (unsigned)__ballot(sup);
        if ((((supb >> l) & 1u) == 0u) && (((validMask >> l) & 1u) != 0u)) {
          activeMask |= (1u << l);
          float ox1 = __shfl(X1, l), oy1 = __shfl(Y1, l);
          float ox2 = __shfl(X2, l), oy2 = __shfl(Y2, l);
          float oar = __shfl(AR, l);
          if (lane > l) {                            // forward-only suppression
            float xx1 = fmaxf(ox1, X1), yy1 = fmaxf(oy1, Y1);
            float xx2 = fminf(ox2, X2), yy2 = fminf(oy2, Y2);
            float ww = fmaxf(xx2 - xx1 + 1.0f, 0.0f);
            float hh = fmaxf(yy2 - yy1 + 1.0f, 0.0f);
            float inter = ww * hh;
            float iou = inter / (oar + AR - inter);
            if (iou > NMS_THf) sup = true;
          }
        }
      }
      if (((activeMask >> lane) & 1u) != 0u) {
        int id = sidx[i];
        if (id < NDET) keep[id] = 1;                 // keep[] pre-zeroed each call
      }
      if (lane == 0) sActive = activeMask;
    }
    __syncthreads();

    // --- whole block: OR this group's kept boxes into all later mask words ---
    unsigned am = sActive;
    if (am) {
      for (int w = g + 1 + tid; w < NGRP; w += blockDim.x) {
        unsigned cur = mask[w];
        if (cur == 0xFFFFFFFFu) continue;            // word fully suppressed already
        int jbase = w * 32;
        __builtin_prefetch(bb + jbase, 0, 1);        // -> global_prefetch_b8
        unsigned add = 0u;
        for (int b = 0; b < 32; ++b) {
          if ((cur >> b) & 1u) continue;             // already suppressed: skip
          int j = jbase + b;
          float4 J = bb[j];                          // global_load_b128 (L2-hot)
          float jar = bar[j];
          unsigned m = am;
          bool supj = false;
          while (m) {
            int k = __ffs(m) - 1; m &= m - 1u;
            float4 K = sbox[k];
            float xx1 = fmaxf(K.x, J.x), yy1 = fmaxf(K.y, J.y);
            float xx2 = fminf(K.z, J.z), yy2 = fminf(K.w, J.w);
            float ww = fmaxf(xx2 - xx1 + 1.0f, 0.0f);
            float hh = fmaxf(yy2 - yy1 + 1.0f, 0.0f);
            float inter = ww * hh;
            float iou = inter / (sar[k] + jar - inter);
            if (iou > NMS_THf) { supj = true; break; }
          }
          if (supj) add |= (1u << b);
        }
        if (add) mask[w] |= add;                     // w owned by exactly one thread
      }
    }
    __syncthreads();
  }
}

// ---------------- 5) finalize: threshold + zero, anchor order ----------------
__global__ void finalize_kernel(const float* __restrict__ dx1, const float* __restrict__ dy1,
                                const float* __restrict__ dx2, const float* __restrict__ dy2,
                                const float* __restrict__ dsc, const int* __restrict__ keep,
                                float* __restrict__ out) {
  int n = blockIdx.x * blockDim.x + threadIdx.x;
  if (n >= NDET) return;
  float s = dsc[n];
  float f = (keep[n] != 0 && s > FIN_THf) ? 1.0f : 0.0f;
  out[5 * n + 0] = dx1[n] * f;
  out[5 * n + 1] = dy1[n] * f;
  out[5 * n + 2] = dx2[n] * f;
  out[5 * n + 3] = dy2[n] * f;
  out[5 * n + 4] = s * f;
}

extern "C" void kernel_launch(void* const* d_in, const int* in_sizes, int n_in,
                              void* d_out, int out_size, void* d_ws, size_t ws_size,
                              hipStream_t stream) {
  (void)in_sizes; (void)n_in; (void)out_size; (void)ws_size;

  // workspace layout (~1.25 MB); all offsets 16B-aligned
  char* w = (char*)d_ws;
  float* dx1 = (float*)w; w += NPAD * 4;
  float* dy1 = (float*)w; w += NPAD * 4;
  float* dx2 = (float*)w; w += NPAD * 4;
  float* dy2 = (float*)w; w += NPAD * 4;
  float* dsc = (float*)w; w += NPAD * 4;
  int*   keep = (int*)w;  w += NPAD * 4;
  float* key = (float*)w; w += MSORT * 4;
  int*   idx = (int*)w;   w += MSORT * 4;
  float4* bb = (float4*)w; w += NPAD * 16;
  float* bar = (float*)w; w += NPAD * 4;
  float* bvl = (float*)w; w += NPAD * 4;

  InPtrs in;
  for (int i = 0; i < 12; ++i) in.p[i] = (const float*)d_in[i];

  decode_init_kernel<<<MSORT / 256, 256, 0, stream>>>(in, dx1, dy1, dx2, dy2, dsc, keep, key, idx);

  // Bitonic sort: 15 launches total (vs 120 flat).
  bitonic_local_full_kernel<<<MSORT / TILE, STHREADS, 0, stream>>>(key, idx);
  for (int k = TILE * 2; k <= MSORT; k <<= 1) {
    for (int j = k >> 1; j >= TILE; j >>= 1)
      bitonic_step_kernel<<<MSORT / 256, 256, 0, stream>>>(key, idx, j, k);
    bitonic_local_merge_kernel<<<MSORT / TILE, STHREADS, 0, stream>>>(key, idx, k);
  }

  gather_kernel<<<(NPAD + 255) / 256, 256, 0, stream>>>(dx1, dy1, dx2, dy2, key, idx,
                                                        bb, bar, bvl);

  nms_kernel<<<1, 1024, 0, stream>>>(bb, bar, bvl, idx, keep);

  finalize_kernel<<<(NDET + 255) / 256, 256, 0, stream>>>(dx1, dy1, dx2, dy2, dsc, keep,
                                                          (float*)d_out);
}